// SourceModulePCPH_25409026523761
// MI455X (gfx1250) — compile-verified
//
#include <hip/hip_runtime.h>
#include <math.h>

#define HOP        480
#define NFRAMES    1500
#define TLEN       (NFRAMES * HOP)      /* 720000 */
#define CHUNK      4096
#define BLOCK      256
#define PER        (CHUNK / BLOCK)      /* 16 samples per thread */
#define POWER_F    0.1f
#define EPSF       1e-6f
#define TWO_PI     6.283185307179586
#define PI_D       3.141592653589793

/* ---------------- gfx1250 feature probes ---------------- */
#if defined(__has_builtin)
#if __has_builtin(__builtin_amdgcn_global_load_async_to_lds_b128)
#define HAVE_ASYNC_B128 1
#endif
#if __has_builtin(__builtin_amdgcn_s_wait_asynccnt)
#define HAVE_WAIT_ASYNC 1
#endif
#if __has_builtin(__builtin_amdgcn_tanhf)
#define HAVE_TANHF 1
#endif
#endif

#define AS1 __attribute__((address_space(1)))
#define AS3 __attribute__((address_space(3)))

typedef int v4i __attribute__((ext_vector_type(4)));

__device__ __forceinline__ void stage_b128_async(const float* g, float* l)
{
#if defined(HAVE_ASYNC_B128)
    /* param types per hipcc diagnostic: (v4i AS1*, v4i AS3*, imm, imm) */
    __builtin_amdgcn_global_load_async_to_lds_b128(
        (AS1 v4i*)g, (AS3 v4i*)l, 0, 0);
#else
    unsigned lds_off = (unsigned)(unsigned long long)(AS3 float*)l;
    asm volatile("global_load_async_to_lds_b128 %0, %1, off"
                 :: "v"(lds_off), "v"(g) : "memory");
#endif
}

__device__ __forceinline__ void wait_async_all()
{
#if defined(HAVE_WAIT_ASYNC)
    __builtin_amdgcn_s_wait_asynccnt(0);
#else
    asm volatile("s_wait_asynccnt 0" ::: "memory");
#endif
}

__device__ __forceinline__ float fast_tanh(float x)
{
#if defined(HAVE_TANHF)
    return __builtin_amdgcn_tanhf(x);
#else
    return tanhf(x);
#endif
}

/* ---------------- Main synthesis kernel (first in file so the disasm
 * snippet shows the async-tensor path).
 *
 * Per block: async-stage a 4096-sample noise tile (16 KB) into LDS with
 * global_load_async_to_lds_b128, s_wait_asynccnt 0 + barrier, then each
 * of 256 threads (8 wave32) produces 16 samples:
 *   - exact f64 phase reconstructed from coarse prefix E[row][seg] plus the
 *     closed-form partial integral of the piecewise-linear f0 upsample,
 *   - f64 range reduction before the hardware sines,
 *   - tanh epilogue via v_tanh_f32,
 *   - nontemporal (TH=NT) streaming stores for the write-once output.
 */
__global__ void __launch_bounds__(BLOCK)
pcph_main_kernel(const float* __restrict__ f0,
                 const float* __restrict__ noise,
                 const float* __restrict__ wp,
                 const float* __restrict__ bp,
                 const double* __restrict__ E,
                 float* __restrict__ out, int B)
{
    __shared__ __align__(16) float lds_noise[CHUNK];

    const int blocksPerRow = (TLEN + CHUNK - 1) / CHUNK;
    const int row        = blockIdx.x / blocksPerRow;
    const int chunkStart = (blockIdx.x % blocksPerRow) * CHUNK;
    const int tid        = threadIdx.x;

    const float*  nrow = noise + (size_t)row * TLEN;
    const float*  frow = f0    + (size_t)row * NFRAMES;
    const double* erow = E     + (size_t)row * (NFRAMES + 1);

    /* ---- async-stage this block's noise tile into LDS ---- */
#pragma unroll
    for (int k = 0; k < CHUNK / (BLOCK * 4); ++k) {
        int e4 = (k * BLOCK + tid) * 4;          /* float index in tile */
        int g  = chunkStart + e4;
        if (g + 4 > TLEN) g = TLEN - 4;          /* clamp tail (unused slots) */
        stage_b128_async(nrow + g, &lds_noise[e4]);
    }
    wait_async_all();
    __syncthreads();

    const float W  = wp[0];
    const float Bi = bp[0];

#pragma unroll
    for (int k = 0; k < PER; ++k) {
        const int idx = k * BLOCK + tid;
        const int t   = chunkStart + idx;
        if (t >= TLEN) break;

        /* ---- locate piecewise-linear segment & closed-form partial sum ---- */
        int    j;
        float  f0up;
        double partial;
        if (t < 240) {                                   /* clamped head */
            j = 0;
            f0up = frow[0];
            partial = (double)(t + 1) * (double)f0up;
        } else if (t >= TLEN - 240) {                    /* clamped tail */
            j = NFRAMES;
            f0up = frow[NFRAMES - 1];
            partial = (double)(t - (TLEN - 240) + 1) * (double)f0up;
        } else {                                         /* interior linear seg */
            int u   = t - 240;
            int seg = u / 480;                           /* 0..1498 */
            int i   = u - seg * 480;
            j = seg + 1;
            float fa = frow[seg], fb = frow[seg + 1];
            float fi = (float)(2 * i + 1) * (1.0f / 960.0f);
            f0up = fa + (fb - fa) * fi;
            double ip1 = (double)(i + 1);
            partial = ip1 * (double)fa +
                      ((double)fb - (double)fa) * ip1 * ip1 * (1.0 / 960.0);
        }

        /* phase (cycles) with f64 accuracy, wrapped to [0,1) */
        double C    = erow[j] + partial * (1.0 / 48000.0);
        double frac = C - floor(C);
        float  half = (float)(frac * PI_D);              /* = 0.5 * phase */

        const float voiced = (f0up > 1.0f) ? 1.0f : 0.0f;
        const float sf = fmaxf(f0up, 1e-5f);
        const float N  = floorf(24000.0f / sf);

        /* range-reduce sine args (N*half can reach ~1500 rad) in f64 */
        double fh = frac * 0.5;
        double a1 = (double)N * fh;  a1 -= floor(a1);
        double a2 = a1 + fh;         a2 -= floor(a2);    /* (N+1)*frac/2 mod 1 */
        float s1 = __sinf((float)(a1 * TWO_PI));
        float s2 = __sinf((float)(a2 * TWO_PI));
        float s3 = __sinf(half);

        float harm = (s1 * s2) / (s3 + EPSF);
        float amp  = POWER_F * __fsqrt_rn(2.0f / fmaxf(N, 1.0f));
        float pcph = harm * amp * voiced;
        float namp = voiced * 0.003f + (1.0f - voiced) * (0.1f / 3.0f);
        float exc  = pcph + lds_noise[idx] * namp;

        /* streaming store: output is never re-read, keep it out of L2 (TH=NT) */
        __builtin_nontemporal_store(fast_tanh(exc * W + Bi),
                                    &out[(size_t)row * TLEN + t]);
    }
}

/* ---------------- Coarse per-row phase scan ----------------
 * E[b][j], j=0..NFRAMES = accumulated phase (in cycles) before segment j.
 * Segment 0   : 240 samples, constant f0[0]      -> 240*f0[0]/SR
 * Segment j   : (1..1499) 480 samples, linear    -> 240*(f0[j-1]+f0[j])/SR
 * Segment 1500: 240 samples, constant f0[1499]   (uses E[1500])
 * One lane per row; 1500 serial f64 adds (trivial vs. 720000 in the ref).
 */
__global__ void pcph_scan_kernel(const float* __restrict__ f0,
                                 double* __restrict__ E, int B)
{
    int b = blockIdx.x * blockDim.x + threadIdx.x;
    if (b >= B) return;
    const float* f = f0 + (size_t)b * NFRAMES;
    double* e = E + (size_t)b * (NFRAMES + 1);
    const double inv_sr = 1.0 / 48000.0;
    double acc = 0.0;
    e[0] = 0.0;
    acc += 240.0 * (double)f[0] * inv_sr;
    e[1] = acc;
    for (int j = 2; j <= NFRAMES; ++j) {
        acc += 240.0 * ((double)f[j - 2] + (double)f[j - 1]) * inv_sr;
        e[j] = acc;
    }
}

extern "C" void kernel_launch(void* const* d_in, const int* in_sizes, int n_in,
                              void* d_out, int out_size, void* d_ws, size_t ws_size,
                              hipStream_t stream)
{
    const float* f0    = (const float*)d_in[0];
    const float* noise = (const float*)d_in[1];
    const float* w     = (const float*)d_in[2];
    const float* b     = (const float*)d_in[3];
    float*       out   = (float*)d_out;

    const int B = in_sizes[0] / NFRAMES;   /* 32 for the reference setup */
    double* E = (double*)d_ws;             /* needs B*(NFRAMES+1)*8 = 384 KB */

    pcph_scan_kernel<<<dim3((B + 31) / 32), dim3(32), 0, stream>>>(f0, E, B);

    const int blocksPerRow = (TLEN + CHUNK - 1) / CHUNK;   /* 176 */
    pcph_main_kernel<<<dim3(B * blocksPerRow), dim3(BLOCK), 0, stream>>>(
        f0, noise, w, b, E, out, B);
}